// BoxProcessor_8641474200314
// MI455X (gfx1250) — compile-verified
//
#include <hip/hip_runtime.h>

// YOLO box decode on MI455X: pure HBM stream (209 MB in + 17 MB out @ 23.3 TB/s
// ~ 9.7 us floor; AI ~ 0.7 FLOP/B -> WMMA inapplicable). CDNA5 path used:
// async global->LDS b128 pipeline (ASYNCcnt) with SADDR addressing and NT
// cache policy; direct NT 128-bit loads for box channels on LOADcnt in parallel.

#define NB   32            // batch
#define NA   3             // anchors
#define NV   85            // channels per anchor
#define HW   6400          // 80*80
#define WDIM 80
#define NCLS 80
#define CHUNK 8            // class channels per async chunk
#define NCHUNK (NCLS / CHUNK)     // 10
#define TPB  320           // 10 waves
#define CPT  4             // cells per thread (b128)
#define CPB  (TPB * CPT)   // 1280 cells per block; 6400/1280 = 5 blocks exact

#define BOXES_ELEMS ((size_t)NB * NA * HW * 6)   // 3,686,400

typedef __attribute__((ext_vector_type(4))) float v4f;   // native vector: OK for
                                                         // nontemporal builtins

// Async b128 global->LDS, SADDR form: per-lane LDS byte offset (VGPR),
// per-lane 32-bit byte offset (VGPR), uniform 64-bit base (SGPR pair).
// NT hint: input is streamed once (209 MB > 192 MB L2).
__device__ __forceinline__ void async_ld_b128(const float* sbase, unsigned voff,
                                              unsigned lds_off) {
    asm volatile("global_load_async_to_lds_b128 %0, %1, %2 th:TH_LOAD_NT"
                 :: "v"(lds_off), "v"(voff), "s"(sbase)
                 : "memory");
}

// Wait until ASYNCcnt <= n (async loads retire in issue order).
#define WAIT_ASYNC_LE(n) asm volatile("s_wait_asynccnt " #n ::: "memory")

__device__ __forceinline__ unsigned lds_off_of(const void* p) {
    // Flat shared address = {aperture_hi32, lds_offset32}; low word is the
    // wave-relative LDS byte offset the async instruction consumes.
    return (unsigned)(unsigned long long)p;
}

__device__ __forceinline__ float fast_sigmoid(float t) {
    return 1.0f / (1.0f + __expf(-t));
}

extern "C" __global__ void __launch_bounds__(TPB)
yolo_decode_kernel(const float* __restrict__ x,
                   const float* __restrict__ anchors,
                   float* __restrict__ boxes,
                   float* __restrict__ mask)
{
    __shared__ v4f sCls[2][CHUNK][TPB];      // 2*8*320*16B = 80 KB

    const int tid   = threadIdx.x;
    const int pair  = blockIdx.y;            // 0..95 == n*3 + a
    const int a     = pair % NA;
    const int cell0 = blockIdx.x * CPB + tid * CPT;   // 4 cells, same row

    // Uniform plane base (SGPR); per-lane byte offset within plane (VGPR).
    const float* gb     = x + (size_t)pair * NV * HW;   // n*255*HW + a*85*HW
    const unsigned voff = (unsigned)cell0 * 4u;

    // Kick off class chunk 0 (ASYNCcnt) before touching box channels.
    const float* gcls = gb + (size_t)5 * HW;
#pragma unroll
    for (int c = 0; c < CHUNK; ++c)
        async_ld_b128(gcls + (size_t)c * HW, voff, lds_off_of(&sCls[0][c][tid]));

    // Box channels: direct NT 128-bit loads (LOADcnt, overlaps async pipe).
    const v4f tx4 = __builtin_nontemporal_load((const v4f*)(gb + 0 * HW + cell0));
    const v4f ty4 = __builtin_nontemporal_load((const v4f*)(gb + 1 * HW + cell0));
    const v4f tw4 = __builtin_nontemporal_load((const v4f*)(gb + 2 * HW + cell0));
    const v4f th4 = __builtin_nontemporal_load((const v4f*)(gb + 3 * HW + cell0));
    const v4f ob4 = __builtin_nontemporal_load((const v4f*)(gb + 4 * HW + cell0));

    const float aw  = anchors[a * 2 + 0];
    const float ah  = anchors[a * 2 + 1];
    const float gy  = (float)(cell0 / WDIM);
    const float gx0 = (float)(cell0 % WDIM);

    float obj[CPT], bx[CPT], by[CPT], bw[CPT], bh[CPT];
#pragma unroll
    for (int j = 0; j < CPT; ++j) {
        obj[j] = ob4[j];                     // raw, no sigmoid (matches ref)
        bx[j]  = fast_sigmoid(tx4[j]) + (gx0 + (float)j);
        by[j]  = fast_sigmoid(ty4[j]) + gy;
        bw[j]  = __expf(tw4[j]) * aw;
        bh[j]  = __expf(th4[j]) * ah;
    }

    // Class argmax: strict '>' ascending == jnp.argmax first-occurrence ties.
    float best[CPT];
    int   bcls[CPT];
#pragma unroll
    for (int j = 0; j < CPT; ++j) { best[j] = -__builtin_inff(); bcls[j] = 0; }

#pragma unroll
    for (int k = 0; k < NCHUNK; ++k) {
        if (k + 1 < NCHUNK) {
            const float* gn = gcls + (size_t)(k + 1) * CHUNK * HW;
#pragma unroll
            for (int c = 0; c < CHUNK; ++c)
                async_ld_b128(gn + (size_t)c * HW, voff,
                              lds_off_of(&sCls[(k + 1) & 1][c][tid]));
            WAIT_ASYNC_LE(8);    // previous chunk fully retired (in-order)
        } else {
            WAIT_ASYNC_LE(0);    // drain final chunk
        }
#pragma unroll
        for (int c = 0; c < CHUNK; ++c) {
            const v4f q = sCls[k & 1][c][tid];   // 4 cells of class (k*8+c)
            const int idx = k * CHUNK + c;
#pragma unroll
            for (int j = 0; j < CPT; ++j) {
                const float s = q[j] * obj[j];
                if (s > best[j]) { best[j] = s; bcls[j] = idx; }
            }
        }
    }

    // Outputs: boxes (N,A,H,W,6) then mask (N,A,H,W) as 0/1 float. Four cells
    // per thread -> 24 contiguous floats (96B aligned) -> 6 NT b128 stores.
    const size_t ci0 = (size_t)pair * HW + (size_t)cell0;
    v4f* bo = (v4f*)(boxes + ci0 * 6);
    __builtin_nontemporal_store((v4f){bx[0], by[0], bw[0], bh[0]},                 bo + 0);
    __builtin_nontemporal_store((v4f){best[0], (float)bcls[0], bx[1], by[1]},      bo + 1);
    __builtin_nontemporal_store((v4f){bw[1], bh[1], best[1], (float)bcls[1]},      bo + 2);
    __builtin_nontemporal_store((v4f){bx[2], by[2], bw[2], bh[2]},                 bo + 3);
    __builtin_nontemporal_store((v4f){best[2], (float)bcls[2], bx[3], by[3]},      bo + 4);
    __builtin_nontemporal_store((v4f){bw[3], bh[3], best[3], (float)bcls[3]},      bo + 5);

    const v4f m = { best[0] > 0.5f ? 1.0f : 0.0f,
                    best[1] > 0.5f ? 1.0f : 0.0f,
                    best[2] > 0.5f ? 1.0f : 0.0f,
                    best[3] > 0.5f ? 1.0f : 0.0f };
    __builtin_nontemporal_store(m, (v4f*)(mask + ci0));
}

extern "C" void kernel_launch(void* const* d_in, const int* in_sizes, int n_in,
                              void* d_out, int out_size, void* d_ws, size_t ws_size,
                              hipStream_t stream) {
    const float* x       = (const float*)d_in[0];
    const float* anchors = (const float*)d_in[1];
    float* boxes = (float*)d_out;
    float* mask  = boxes + BOXES_ELEMS;

    dim3 grid(HW / CPB, NB * NA);            // (5, 96)
    hipLaunchKernelGGL(yolo_decode_kernel, grid, dim3(TPB), 0, stream,
                       x, anchors, boxes, mask);
}